// LocalSpectralAdapter_3882650437189
// MI455X (gfx1250) — compile-verified
//
#include <hip/hip_runtime.h>
#include <math.h>

typedef __attribute__((ext_vector_type(2))) float v2f;
typedef __attribute__((ext_vector_type(8))) float v8f;

#define T_LEN 1024
#define V_LEN 512
#define B_LEN 64
#define NBINS 16

__constant__ int KBINS[NBINS] = {1,2,3,4,5,6,7,8,12,16,24,32,48,64,96,128};

// Workspace layout (float offsets)
#define WS_BPROJ 0                          // 32 x 1024   projection basis (cos rows, -sin rows)
#define WS_WREC  (32 * 1024)                // 1024 x 64   reconstruction basis with fade folded in
#define WS_Q     (WS_WREC + 1024 * 64)      // 64 x 64 x 512  combined gain coefficients

// ---------------------------------------------------------------------------
// Kernel 0: build basis tables. theta reduced mod 2*pi exactly via integer mod.
// ---------------------------------------------------------------------------
__global__ void lsa_build_tables(float* __restrict__ ws) {
    int t = blockIdx.x * blockDim.x + threadIdx.x;
    if (t >= T_LEN) return;
    // early weight: CW = max(4, 1024/20) = 51; FS = 512-25 = 487; FE = 512+25 = 537
    const float FS = 487.0f, FE = 537.0f;
    float tf = (float)t;
    float ew = (tf < FS) ? 1.0f : ((tf < FE) ? (1.0f - (tf - FS) / (FE - FS)) : 0.0f);
    float lw = 1.0f - ew;

    float* Bp = ws + WS_BPROJ;
    float* Wr = ws + WS_WREC;
    const float STEP = 6.283185307179586f / (float)T_LEN;
#pragma unroll
    for (int j = 0; j < NBINS; ++j) {
        int m = (KBINS[j] * t) & (T_LEN - 1);     // exact phase mod 2*pi
        float th = (float)m * STEP;
        float s, c;
        sincosf(th, &s, &c);
        // projection: Xr = sum cos * x ; Xi = sum (-sin) * x   (e^{-i theta})
        Bp[j * T_LEN + t]          = c;
        Bp[(16 + j) * T_LEN + t]   = -s;
        // reconstruction: Re(P * e^{+i theta}) = P_r cos - P_i sin, fade folded in
        Wr[t * 64 + j]             = ew * c;
        Wr[t * 64 + 16 + j]        = -ew * s;
        Wr[t * 64 + 32 + j]        = lw * c;
        Wr[t * 64 + 48 + j]        = -lw * s;
    }
}

// ---------------------------------------------------------------------------
// Kernel 1: fused projection + gains.
// Each wave owns one 16-wide v tile and BOTH m-tiles (Xr rows 0..15, Xi rows
// 16..31), sharing the x B-fragment between two WMMAs per K-step. After the
// K-loop, lane element r of accR/accI = (Xr, Xi) at the SAME (j, v), so the
// complex gain multiply is lane-local and Q is written directly (no P buffer).
// Q rows: [0..15]=Pe_r [16..31]=Pe_i [32..47]=Pl_r [48..63]=Pl_i, scaled 2/T.
// ---------------------------------------------------------------------------
__global__ void lsa_proj_gains(const float* __restrict__ x,
                               const float* __restrict__ ger, const float* __restrict__ gei,
                               const float* __restrict__ glr, const float* __restrict__ gli,
                               float* __restrict__ ws) {
    const float* Bp = ws + WS_BPROJ;

    const int lane = threadIdx.x & 31;
    const int wave = threadIdx.x >> 5;
    const int nt   = blockIdx.x * 8 + wave;    // 0..31 v-tile
    const int b    = blockIdx.y;

    const int half = lane >> 4;                // 0: K{0,1}, 1: K{2,3}
    const int l15  = lane & 15;
    const int m    = l15;                      // A-frag row (same for both lane halves)
    const int n    = nt * 16 + l15;            // B-frag column (v)

    const float* xb = x + (size_t)b * T_LEN * V_LEN;

    v8f accR = {};   // Xr tile (Bproj rows 0..15)
    v8f accI = {};   // Xi tile (Bproj rows 16..31)
    for (int k = 0; k < T_LEN; k += 4) {
        const int ka = k + 2 * half;
        v2f a0, a1, bf;
        a0.x = Bp[m * T_LEN + ka];
        a0.y = Bp[m * T_LEN + ka + 1];
        a1.x = Bp[(16 + m) * T_LEN + ka];
        a1.y = Bp[(16 + m) * T_LEN + ka + 1];
        bf.x = xb[(size_t)ka * V_LEN + n];
        bf.y = xb[(size_t)(ka + 1) * V_LEN + n];
        accR = __builtin_amdgcn_wmma_f32_16x16x4_f32(false, a0, false, bf,
                                                     (short)0, accR, false, false);
        accI = __builtin_amdgcn_wmma_f32_16x16x4_f32(false, a1, false, bf,
                                                     (short)0, accI, false, false);
    }

    // Fused gain epilogue: lane element r -> (j = r + 8*half, v = n)
    float* Qb = ws + WS_Q + (size_t)b * 64 * V_LEN;
    const float sc = 2.0f / (float)T_LEN;
#pragma unroll
    for (int r = 0; r < 8; ++r) {
        const int j  = r + 8 * half;           // C/D layout: M = r + 8*(lane>=16)
        const float xr = accR[r];
        const float xi = accI[r];
        const float er = ger[n * NBINS + j], ei = gei[n * NBINS + j];
        const float lr = glr[n * NBINS + j], li = gli[n * NBINS + j];
        Qb[j * V_LEN + n]        = sc * (xr * er - xi * ei);   // Pe_r
        Qb[(16 + j) * V_LEN + n] = sc * (xr * ei + xi * er);   // Pe_i
        Qb[(32 + j) * V_LEN + n] = sc * (xr * lr - xi * li);   // Pl_r
        Qb[(48 + j) * V_LEN + n] = sc * (xr * li + xi * lr);   // Pl_i
    }
}

// ---------------------------------------------------------------------------
// Kernel 2: reconstruction GEMM + residual add
// out_b(1024x512) = x_b + Wrec(1024x64) @ Q_b(64x512)
// ---------------------------------------------------------------------------
__global__ void lsa_recon(const float* __restrict__ x, const float* __restrict__ ws,
                          float* __restrict__ out) {
    const float* Wr = ws + WS_WREC;

    const int lane = threadIdx.x & 31;
    const int wave = threadIdx.x >> 5;
    const int tt = blockIdx.x * 8 + wave;      // 0..63 t-tile
    const int vt = blockIdx.y;                 // 0..31 v-tile
    const int b  = blockIdx.z;

    const int half = lane >> 4;
    const int l15  = lane & 15;
    const int trow = tt * 16 + l15;
    const int n    = vt * 16 + l15;

    const float* Qb = ws + WS_Q + (size_t)b * 64 * V_LEN;

    v8f acc = {};
#pragma unroll
    for (int k = 0; k < 64; k += 4) {
        const int ka = k + 2 * half;
        v2f a, bf;
        a.x  = Wr[trow * 64 + ka];
        a.y  = Wr[trow * 64 + ka + 1];
        bf.x = Qb[ka * V_LEN + n];
        bf.y = Qb[(ka + 1) * V_LEN + n];
        acc = __builtin_amdgcn_wmma_f32_16x16x4_f32(false, a, false, bf,
                                                    (short)0, acc, false, false);
    }

    const float* xb = x + (size_t)b * T_LEN * V_LEN;
    float* ob       = out + (size_t)b * T_LEN * V_LEN;
#pragma unroll
    for (int r = 0; r < 8; ++r) {
        const int row = tt * 16 + r + 8 * half;
        const size_t off = (size_t)row * V_LEN + vt * 16 + l15;
        ob[off] = xb[off] + acc[r];
    }
}

extern "C" void kernel_launch(void* const* d_in, const int* in_sizes, int n_in,
                              void* d_out, int out_size, void* d_ws, size_t ws_size,
                              hipStream_t stream) {
    const float* x   = (const float*)d_in[0];
    const float* ger = (const float*)d_in[1];
    const float* gei = (const float*)d_in[2];
    const float* glr = (const float*)d_in[3];
    const float* gli = (const float*)d_in[4];
    float* out = (float*)d_out;
    float* ws  = (float*)d_ws;

    // 0) basis tables (Bproj 128KB, Wrec 256KB)
    lsa_build_tables<<<dim3(4), 256, 0, stream>>>(ws);

    // 1) fused projection + gains: per b, 32 v-tiles -> 4 blocks of 8 waves
    lsa_proj_gains<<<dim3(4, B_LEN), 256, 0, stream>>>(x, ger, gei, glr, gli, ws);

    // 2) reconstruction + residual add
    lsa_recon<<<dim3(8, V_LEN / 16, B_LEN), 256, 0, stream>>>(x, ws, out);
}